// GaborLayer_13245679141131
// MI455X (gfx1250) — compile-verified
//
#include <hip/hip_runtime.h>
#include <math.h>

typedef __attribute__((ext_vector_type(2))) float v2f;
typedef __attribute__((ext_vector_type(8))) float v8f;

#define N_GABORS 256
#define HW (512 * 512)
#define TWO_PI_F 6.28318530717958647692f

__device__ __forceinline__ float clampf(float x, float lo, float hi) {
    return fminf(fmaxf(x, lo), hi);
}

// One wave handles a 16-pixel tile x 16 output columns (3 channels used).
// K dimension = 512 terms (256 gabors x {cos-part, sin-part}), stepped 4 at a
// a time by V_WMMA_F32_16X16X4_F32. Lanes 0-15 evaluate gabor 2k, lanes 16-31
// gabor 2k+1 -> exactly one exp/cos/sin triple per lane per iteration.
__global__ __launch_bounds__(256) void gabor_wmma_kernel(
    const float* __restrict__ grid_x, const float* __restrict__ grid_y,
    const float* __restrict__ u,      const float* __restrict__ v,
    const float* __restrict__ theta,  const float* __restrict__ rel_sigma,
    const float* __restrict__ rel_freq, const float* __restrict__ gamma,
    const float* __restrict__ psi,    const float* __restrict__ amplitude,
    float* __restrict__ out)
{
    // Per-gabor derived params: uc, vc, cos(th), sin(th), 1/(2*sig^2),
    // 1/(2*gam^2), freq, pad  -> 8 KB
    __shared__ float sP[N_GABORS * 8];
    // B coefficients, padded to 4 channels (slot 3 == 0 feeds columns 3..15)
    __shared__ float sBc[N_GABORS * 4];   // amp*cos(phase)
    __shared__ float sBs[N_GABORS * 4];   // -amp*sin(phase)

    const int t = threadIdx.x;  // 256 threads <-> 256 gabors
    {
        float uc = clampf(u[t], -1.f, 1.f);
        float vc = clampf(v[t], -1.f, 1.f);
        float th = clampf(theta[t], -2.f, 2.f) * TWO_PI_F;
        float sg = clampf(rel_sigma[t], 1e-5f, 5.f);
        float gm = clampf(gamma[t],     1e-5f, 5.f);
        float cr = __cosf(th), sr = __sinf(th);
        float i2s = 1.f / (2.f * sg * sg);
        float i2g = 1.f / (2.f * gm * gm);
        float fr  = TWO_PI_F * __expf(-rel_freq[t]);
        sP[t * 8 + 0] = uc;  sP[t * 8 + 1] = vc;
        sP[t * 8 + 2] = cr;  sP[t * 8 + 3] = sr;
        sP[t * 8 + 4] = i2s; sP[t * 8 + 5] = i2g;
        sP[t * 8 + 6] = fr;  sP[t * 8 + 7] = 0.f;
        #pragma unroll
        for (int c = 0; c < 4; ++c) {
            float bc = 0.f, bs = 0.f;
            if (c < 3) {
                float ph = clampf(psi[t * 3 + c], -1.f, 1.f) * TWO_PI_F;
                float a  = clampf(amplitude[t * 3 + c], 0.f, 1.f);
                bc =  a * __cosf(ph);
                bs = -a * __sinf(ph);
            }
            sBc[t * 4 + c] = bc;
            sBs[t * 4 + c] = bs;
        }
    }
    __syncthreads();

    const int lane = threadIdx.x & 31;
    const int wave = threadIdx.x >> 5;
    const int half = lane >> 4;        // 0: gabor 2k (K=0,1)  1: gabor 2k+1 (K=2,3)
    const int col  = lane & 15;        // A: pixel row M; B: channel column N
    const int pixBase = (blockIdx.x * 8 + wave) * 16;
    const int myPix   = pixBase + col;

    const float px = grid_x[myPix];
    const float py = grid_y[myPix];
    const int colc = (col < 3) ? col : 3;   // slot 3 holds 0.0f

    v8f acc = {};

    #pragma unroll 4
    for (int k = 0; k < N_GABORS / 2; ++k) {
        const int g = 2 * k + half;
        const float* p = &sP[g * 8];
        float dx = px - p[0];
        float dy = py - p[1];
        float cr = p[2], sr = p[3];
        float xr = dx * cr + dy * sr;
        float yr = dy * cr - dx * sr;
        float e  = -(xr * xr * p[4] + yr * yr * p[5]);
        e = clampf(e, -80.f, 80.f);
        float gau = __expf(e);
        float fx  = p[6] * xr;
        float sf, cf;
        __sincosf(fx, &sf, &cf);

        v2f A; A.x = gau * cf; A.y = gau * sf;          // K pair for this lane
        v2f B; B.x = sBc[g * 4 + colc];                 // cos-term coeff
               B.y = sBs[g * 4 + colc];                 // sin-term coeff (neg folded)

        acc = __builtin_amdgcn_wmma_f32_16x16x4_f32(
            /*neg_a=*/false, A, /*neg_b=*/false, B,
            /*c_mod=*/(short)0, acc, /*reuse_a=*/false, /*reuse_b=*/false);
    }

    // D layout: lane column = channel, VGPR r = pixel row r + 8*half.
    if (col < 3) {
        float* o = out + col * HW + pixBase + half * 8;
        #pragma unroll
        for (int r = 0; r < 8; ++r)
            o[r] = clampf(acc[r], -1.f, 1.f);
    }
}

extern "C" void kernel_launch(void* const* d_in, const int* in_sizes, int n_in,
                              void* d_out, int out_size, void* d_ws, size_t ws_size,
                              hipStream_t stream) {
    const float* grid_x    = (const float*)d_in[0];
    const float* grid_y    = (const float*)d_in[1];
    const float* u         = (const float*)d_in[2];
    const float* v         = (const float*)d_in[3];
    const float* theta     = (const float*)d_in[4];
    const float* rel_sigma = (const float*)d_in[5];
    const float* rel_freq  = (const float*)d_in[6];
    const float* gamma     = (const float*)d_in[7];
    const float* psi       = (const float*)d_in[8];
    const float* amplitude = (const float*)d_in[9];
    float* out = (float*)d_out;

    // 262144 pixels / (8 waves * 16 pixels per wave) = 2048 blocks, exact fit.
    dim3 grid(HW / (8 * 16));
    dim3 block(256);
    gabor_wmma_kernel<<<grid, block, 0, stream>>>(
        grid_x, grid_y, u, v, theta, rel_sigma, rel_freq, gamma, psi, amplitude,
        out);
}